// GCNModel_44667659878955
// MI455X (gfx1250) — compile-verified
//
#include <hip/hip_runtime.h>

typedef __attribute__((ext_vector_type(2))) float v2f;
typedef __attribute__((ext_vector_type(8))) float v8f;

// ---------------------------------------------------------------- degree prep
__global__ void init_deg_kernel(float* __restrict__ deg, int n) {
    int i = blockIdx.x * blockDim.x + threadIdx.x;
    if (i < n) deg[i] = 1.0f;  // self-loop
}

__global__ void count_deg_kernel(const int* __restrict__ dst,
                                 float* __restrict__ deg, int E) {
    int e = blockIdx.x * blockDim.x + threadIdx.x;
    if (e < E) atomicAdd(&deg[dst[e]], 1.0f);
}

__global__ void isqrt_kernel(float* __restrict__ deg, int n) {
    int i = blockIdx.x * blockDim.x + threadIdx.x;
    if (i < n) deg[i] = rsqrtf(deg[i]);
}

// ---------------------------------------------------------------- WMMA GEMM
// One wave computes a 16x64 slab of  H = A' @ W  (W is Kx64), then stores
// ht = H * dis[row] into BOTH ht (gather source) and agg (accumulator init =
// self-loop contribution). When FUSE, the A element is transformed on load:
// A' = relu(A * dis[arow] + bias[k])  (fused finalization of previous layer).
template <bool FUSE>
__global__ void gemm_wmma_kernel(const float* __restrict__ A,
                                 const float* __restrict__ W,
                                 const float* __restrict__ dis,
                                 const float* __restrict__ bias,
                                 float* __restrict__ ht,
                                 float* __restrict__ agg,
                                 int M, int K) {
    const int lane = threadIdx.x & 31;
    const int wave = blockIdx.x * (blockDim.x >> 5) + (threadIdx.x >> 5);
    const int m0 = wave * 16;
    if (m0 >= M) return;                      // whole-wave exit: EXEC stays all-1s

    const int half = lane >> 4;               // 0: K=k0..k0+1, 1: K=k0+2..k0+3
    const int ml   = lane & 15;
    const int arow = m0 + ml;                 // A-matrix row for this lane
    const float dA = FUSE ? dis[arow] : 0.0f;

    v8f c0 = {}, c1 = {}, c2 = {}, c3 = {};

    for (int k0 = 0; k0 < K; k0 += 4) {
        const int ka = k0 + half * 2;
        // A tile 16x4 f32: lane holds A[arow][ka], A[arow][ka+1]
        float a0 = A[(size_t)arow * K + ka];
        float a1 = A[(size_t)arow * K + ka + 1];
        if (FUSE) {
            a0 = fmaxf(fmaf(a0, dA, bias[ka]), 0.0f);
            a1 = fmaxf(fmaf(a1, dA, bias[ka + 1]), 0.0f);
        }
        v2f av; av.x = a0; av.y = a1;

        // B tiles 4x16 f32: v0 = W[ka][n], v1 = W[ka+1][n]
        const float* w0 = W + (size_t)ka * 64;
        const float* w1 = W + (size_t)(ka + 1) * 64;
        v2f b0; b0.x = w0[ml];      b0.y = w1[ml];
        v2f b1; b1.x = w0[16 + ml]; b1.y = w1[16 + ml];
        v2f b2; b2.x = w0[32 + ml]; b2.y = w1[32 + ml];
        v2f b3; b3.x = w0[48 + ml]; b3.y = w1[48 + ml];

        c0 = __builtin_amdgcn_wmma_f32_16x16x4_f32(false, av, false, b0, (short)0, c0, false, false);
        c1 = __builtin_amdgcn_wmma_f32_16x16x4_f32(false, av, false, b1, (short)0, c1, false, false);
        c2 = __builtin_amdgcn_wmma_f32_16x16x4_f32(false, av, false, b2, (short)0, c2, false, false);
        c3 = __builtin_amdgcn_wmma_f32_16x16x4_f32(false, av, false, b3, (short)0, c3, false, false);
    }

    // C/D layout: VGPR r -> row m0 + r + 8*half, column = ml + 16*tile
    for (int r = 0; r < 8; ++r) {
        const int orow = m0 + half * 8 + r;
        const float dr = dis[orow];
        const float v0 = c0[r] * dr;
        const float v1 = c1[r] * dr;
        const float v2 = c2[r] * dr;
        const float v3 = c3[r] * dr;
        float* hp = ht  + (size_t)orow * 64;
        float* ap = agg + (size_t)orow * 64;
        hp[ml] = v0; hp[16 + ml] = v1; hp[32 + ml] = v2; hp[48 + ml] = v3;
        ap[ml] = v0; ap[16 + ml] = v1; ap[32 + ml] = v2; ap[48 + ml] = v3;
    }
}

// ---------------------------------------------------------------- edge scatter
// One wave per edge: gather 64-float row ht[src] (256B coalesced), atomic-add
// into agg[dst]. norm factors are pre-folded into ht/finalization.
__global__ void scatter_edges_kernel(const int* __restrict__ src,
                                     const int* __restrict__ dst,
                                     const float* __restrict__ ht,
                                     float* __restrict__ agg, int E) {
    const int lane = threadIdx.x & 31;
    const int e = blockIdx.x * (blockDim.x >> 5) + (threadIdx.x >> 5);
    if (e >= E) return;
    const int s = src[e];
    const int d = dst[e];
    const float2 v = *(const float2*)(ht + (size_t)s * 64 + lane * 2);
    float* ap = agg + (size_t)d * 64 + lane * 2;
    atomicAdd(ap, v.x);
    atomicAdd(ap + 1, v.y);
}

// ---------------------------------------------------------------- final FC
// out[i] = sum_j relu(agg2[i][j]*dis[i] + b2[j]) * Wfc[j] + bfc
__global__ void fc_out_kernel(const float* __restrict__ agg2,
                              const float* __restrict__ dis,
                              const float* __restrict__ b2,
                              const float* __restrict__ wfc,
                              const float* __restrict__ bfc,
                              float* __restrict__ out, int M) {
    const int lane = threadIdx.x & 31;
    const int i = blockIdx.x * (blockDim.x >> 5) + (threadIdx.x >> 5);
    if (i >= M) return;
    const float di = dis[i];
    const int j = lane * 2;
    const float2 a = *(const float2*)(agg2 + (size_t)i * 64 + j);
    const float h0 = fmaxf(fmaf(a.x, di, b2[j]), 0.0f);
    const float h1 = fmaxf(fmaf(a.y, di, b2[j + 1]), 0.0f);
    float p = h0 * wfc[j] + h1 * wfc[j + 1];
    #pragma unroll
    for (int off = 16; off > 0; off >>= 1) p += __shfl_down(p, off, 32);
    if (lane == 0) out[i] = p + bfc[0];
}

// ---------------------------------------------------------------- launch
extern "C" void kernel_launch(void* const* d_in, const int* in_sizes, int n_in,
                              void* d_out, int out_size, void* d_ws, size_t ws_size,
                              hipStream_t stream) {
    const float* x   = (const float*)d_in[0];
    const int*   ei  = (const int*)d_in[1];
    const float* W1  = (const float*)d_in[2];
    const float* b1  = (const float*)d_in[3];
    const float* W2  = (const float*)d_in[4];
    const float* b2  = (const float*)d_in[5];
    const float* Wfc = (const float*)d_in[6];
    const float* bfc = (const float*)d_in[7];

    const int HID = in_sizes[3];            // 64
    const int IN  = in_sizes[2] / HID;      // 128
    const int N   = in_sizes[0] / IN;       // 100000
    const int E   = in_sizes[1] / 2;        // 1600000
    const int* srcv = ei;
    const int* dstv = ei + E;

    // workspace carve-up
    char* ws = (char*)d_ws;
    float* dis = (float*)ws;                                   // N floats
    size_t off = (((size_t)N * 4) + 255) & ~(size_t)255;
    const size_t rowBytes = (size_t)N * 64 * sizeof(float);
    float* ht   = (float*)(ws + off);                          // gather source
    float* agg1 = (float*)(ws + off + rowBytes);
    float* agg2 = (float*)(ws + off + 2 * rowBytes);

    // degrees -> dis = 1/sqrt(deg)
    init_deg_kernel<<<(N + 255) / 256, 256, 0, stream>>>(dis, N);
    count_deg_kernel<<<(E + 255) / 256, 256, 0, stream>>>(dstv, dis, E);
    isqrt_kernel<<<(N + 255) / 256, 256, 0, stream>>>(dis, N);

    const int gemmBlocks = ((N + 15) / 16 + 7) / 8;   // 8 waves/block, 16 rows/wave
    const int edgeBlocks = (E + 7) / 8;               // 8 waves/block, 1 edge/wave
    const int nodeBlocks = (N + 7) / 8;

    // layer 1: ht = (x@W1)*dis ; agg1 init = ht (self loop) ; scatter edges
    gemm_wmma_kernel<false><<<gemmBlocks, 256, 0, stream>>>(x, W1, dis, nullptr, ht, agg1, N, IN);
    scatter_edges_kernel<<<edgeBlocks, 256, 0, stream>>>(srcv, dstv, ht, agg1, E);

    // layer 2: A = relu(agg1*dis + b1) fused into GEMM load
    gemm_wmma_kernel<true><<<gemmBlocks, 256, 0, stream>>>(agg1, W2, dis, b1, ht, agg2, N, HID);
    scatter_edges_kernel<<<edgeBlocks, 256, 0, stream>>>(srcv, dstv, ht, agg2, E);

    // final fc with fused relu finalization of layer 2
    fc_out_kernel<<<nodeBlocks, 256, 0, stream>>>(agg2, dis, b2, Wfc, bfc, (float*)d_out, N);
}